// SimpleModel_21689584845315
// MI455X (gfx1250) — compile-verified
//
#include <hip/hip_runtime.h>
#include <hip/hip_bf16.h>

// ---------------------------------------------------------------------------
// Bidirectional 5-layer LSTM (B=2048, T=512, H=11) for gfx1250 (MI455X).
// Recurrent + input GEMMs run on V_WMMA_F32_16X16X4_F32 (fp32 matrix pipe).
// One wave = 16 batch rows x 1 direction; full time loop in-wave (no global
// sync). Gate layout padded to 4 N-tiles of 16 so elementwise is lane-local.
// ---------------------------------------------------------------------------

typedef float v2f __attribute__((ext_vector_type(2)));
typedef float v8f __attribute__((ext_vector_type(8)));

#define NB   2048
#define TT   512
#define HH   11
#define GG   44
#define OUTF 22

__device__ __forceinline__ v8f wmma_f32_16x16x4(v2f a, v2f b, v8f c) {
  // (neg_a, A, neg_b, B, c_mod, C, reuse_a, reuse_b)
  return __builtin_amdgcn_wmma_f32_16x16x4_f32(false, a, false, b, (short)0, c,
                                               false, false);
}

__device__ __forceinline__ float fast_sigmoid(float x) {
  return __builtin_amdgcn_rcpf(1.0f + __expf(-x));
}
__device__ __forceinline__ float fast_tanh(float x) {
  // tanh(x) = 1 - 2/(exp(2x)+1); saturates correctly at both ends.
  return 1.0f - 2.0f * __builtin_amdgcn_rcpf(__expf(2.0f * x) + 1.0f);
}
__device__ __forceinline__ v8f splat8(float x) {
  v8f v;
#pragma unroll
  for (int i = 0; i < 8; ++i) v[i] = x;
  return v;
}

// --------------------------- x reduction -----------------------------------
// xsum[b,t] = sum_k x[b,0,k,t];  x is [2048,1,32,512] fp32.
__global__ __launch_bounds__(256) void reduce_x_kernel(
    const float* __restrict__ x, float* __restrict__ xsum) {
  size_t idx = (size_t)blockIdx.x * blockDim.x + threadIdx.x;
  if (idx >= (size_t)NB * TT) return;
  size_t b = idx / TT, t = idx % TT;
  const float* p = x + b * (32u * TT) + t;
  float s = 0.0f;
#pragma unroll
  for (int k = 0; k < 32; ++k) s += p[(size_t)k * TT];
  xsum[idx] = s;
}

// --------------------------- LSTM layer ------------------------------------
// One wave per (16-batch tile, direction). DIN = 1 (layer 0, xsum input) or
// 22 (layers 1..4). Weights per layer passed as base pointers; dir offset
// applied in-kernel (blockIdx.y = direction).
template <int DIN>
__global__ __launch_bounds__(32) void lstm_layer_kernel(
    const float* __restrict__ in, float* __restrict__ out,
    const float* __restrict__ Wih, const float* __restrict__ Whh,
    const float* __restrict__ bih, const float* __restrict__ bhh) {
  constexpr int NSLICE_IN = (DIN + 3) / 4;  // 1 or 6 K-slices of 4
  constexpr int NSLICE_H  = 3;              // K = 11 -> 12

  const int lane  = threadIdx.x;  // wave32
  const int n     = lane & 15;    // WMMA column (hidden / gate index), A row
  const int khalf = lane >> 4;    // 0: K pair {k0,k0+1}; 1: {k0+2,k0+3}
  const int dir   = blockIdx.y;
  const int b0    = blockIdx.x * 16;

  Wih += (size_t)dir * GG * DIN;
  Whh += (size_t)dir * GG * HH;
  bih += (size_t)dir * GG;
  bhh += (size_t)dir * GG;

  // ---- Preload weights in WMMA B-layout (4x16 f32: lanes0-15 K=0,1;
  //      lanes16-31 K=2,3; N = lane%16). Gate gt occupies its own N-tile,
  //      columns 0..10 used, 11..15 zero-padded.
  v2f   Bh[4][NSLICE_H];
  v2f   Bi[4][NSLICE_IN];
  v8f   accBias[4];
  const bool gv = (n < HH);
#pragma unroll
  for (int gt = 0; gt < 4; ++gt) {
    const int g = gt * HH + n;  // row in [G=44] weight matrices
    accBias[gt] = splat8(gv ? (bih[g] + bhh[g]) : 0.0f);
#pragma unroll
    for (int s = 0; s < NSLICE_H; ++s) {
      const int k0 = s * 4 + 2 * khalf;
      float w0 = (gv && k0 + 0 < HH) ? Whh[(size_t)g * HH + k0 + 0] : 0.0f;
      float w1 = (gv && k0 + 1 < HH) ? Whh[(size_t)g * HH + k0 + 1] : 0.0f;
      Bh[gt][s] = v2f{w0, w1};
    }
#pragma unroll
    for (int s = 0; s < NSLICE_IN; ++s) {
      const int k0 = s * 4 + 2 * khalf;
      float w0 = (gv && k0 + 0 < DIN) ? Wih[(size_t)g * DIN + k0 + 0] : 0.0f;
      float w1 = (gv && k0 + 1 < DIN) ? Wih[(size_t)g * DIN + k0 + 1] : 0.0f;
      Bi[gt][s] = v2f{w0, w1};
    }
  }

  // h staging tile for D-layout -> A-layout transpose. 16 rows x 16 cols
  // (cols 12..15 never read; cols 11..15 provably stay 0).
  __shared__ float hbuf[16 * 16];
  for (int i = lane; i < 16 * 16; i += 32) hbuf[i] = 0.0f;
  __syncthreads();  // 1-wave workgroup: S_NOP + compiler fence

  v8f c = splat8(0.0f);

  const size_t inRow = (size_t)(b0 + n) * TT;  // this lane's A-row (batch)
  for (int step = 0; step < TT; ++step) {
    const int t = (dir == 0) ? step : (TT - 1 - step);

    // ---- A (input) slices, loaded straight into WMMA A-layout.
    v2f Ain[NSLICE_IN];
    if (DIN == 1) {
      const float xv = (khalf == 0) ? in[inRow + t] : 0.0f;
      Ain[0] = v2f{xv, 0.0f};
    } else {
      const float* p = in + (inRow + (size_t)t) * DIN;
#pragma unroll
      for (int s = 0; s < NSLICE_IN; ++s) {
        const int k0 = s * 4 + 2 * khalf;  // even -> 8B aligned
        v2f v = v2f{0.0f, 0.0f};
        if (k0 + 1 < DIN) v = *(const v2f*)(p + k0);
        Ain[s] = v;
      }
      // prefetch next timestep's row
      const int tn = (dir == 0) ? (t + 1) : (t - 1);
      if (tn >= 0 && tn < TT)
        __builtin_prefetch(in + (inRow + (size_t)tn) * DIN, 0, 1);
    }

    // ---- A (h state) slices from LDS.
    v2f Ah[NSLICE_H];
#pragma unroll
    for (int s = 0; s < NSLICE_H; ++s) {
      const int k0 = s * 4 + 2 * khalf;
      Ah[s] = *(const v2f*)(&hbuf[(n << 4) + k0]);
    }
    __syncthreads();  // loads of old h ordered before stores of new h

    // ---- Gate pre-activations: g = bias + h@Whh^T + x@Wih^T  (4 N-tiles)
    v8f a0 = accBias[0], a1 = accBias[1], a2 = accBias[2], a3 = accBias[3];
#pragma unroll
    for (int s = 0; s < NSLICE_H; ++s) {
      a0 = wmma_f32_16x16x4(Ah[s], Bh[0][s], a0);
      a1 = wmma_f32_16x16x4(Ah[s], Bh[1][s], a1);
      a2 = wmma_f32_16x16x4(Ah[s], Bh[2][s], a2);
      a3 = wmma_f32_16x16x4(Ah[s], Bh[3][s], a3);
    }
#pragma unroll
    for (int s = 0; s < NSLICE_IN; ++s) {
      a0 = wmma_f32_16x16x4(Ain[s], Bi[0][s], a0);
      a1 = wmma_f32_16x16x4(Ain[s], Bi[1][s], a1);
      a2 = wmma_f32_16x16x4(Ain[s], Bi[2][s], a2);
      a3 = wmma_f32_16x16x4(Ain[s], Bi[3][s], a3);
    }

    // ---- Elementwise cell update. C-layout: VGPR r, lane -> (row r+8*khalf,
    //      col n). i/f/g/o live in the same (r, lane) slot across a0..a3.
    v8f hnew;
#pragma unroll
    for (int r = 0; r < 8; ++r) {
      const float ig = fast_sigmoid(a0[r]);
      const float fg = fast_sigmoid(a1[r]);
      const float gg = fast_tanh(a2[r]);
      const float og = fast_sigmoid(a3[r]);
      const float cr = fg * c[r] + ig * gg;
      c[r]    = cr;
      hnew[r] = og * fast_tanh(cr);
    }

    // ---- Stage new h into LDS (transpose for next step's A operand).
#pragma unroll
    for (int r = 0; r < 8; ++r)
      hbuf[((r + 8 * khalf) << 4) + n] = hnew[r];

    // ---- Emit h to the layer output [B,T,22], features dir*11 + n.
    if (gv) {
#pragma unroll
      for (int r = 0; r < 8; ++r) {
        const int mrow = r + 8 * khalf;
        out[((size_t)(b0 + mrow) * TT + t) * OUTF + (size_t)dir * HH + n] =
            hnew[r];
      }
    }
    __syncthreads();  // new h visible before next iteration's loads
  }
}

// --------------------------- launcher --------------------------------------
extern "C" void kernel_launch(void* const* d_in, const int* in_sizes, int n_in,
                              void* d_out, int out_size, void* d_ws,
                              size_t ws_size, hipStream_t stream) {
  const float* x     = (const float*)d_in[0];  // [2048,1,32,512]
  const float* W_ih0 = (const float*)d_in[1];  // [2,44,1]
  const float* W_ihr = (const float*)d_in[2];  // [4,2,44,22]
  const float* W_hh  = (const float*)d_in[3];  // [5,2,44,11]
  const float* b_ih  = (const float*)d_in[4];  // [5,2,44]
  const float* b_hh  = (const float*)d_in[5];  // [5,2,44]
  float* out = (float*)d_out;                  // [2048,512,22]

  const size_t XS_BYTES  = (size_t)NB * TT * sizeof(float);         // 4 MB
  const size_t ACT_BYTES = (size_t)NB * TT * OUTF * sizeof(float);  // 92 MB
  char*  ws   = (char*)d_ws;
  float* xsum = (float*)(ws);
  float* bufA = (float*)(ws + XS_BYTES);
  float* bufB = (float*)(ws + XS_BYTES + ACT_BYTES);
  (void)ws_size; (void)in_sizes; (void)n_in; (void)out_size;

  // 1) reduce x over the 32-axis.
  {
    const size_t total = (size_t)NB * TT;
    reduce_x_kernel<<<dim3((unsigned)((total + 255) / 256)), dim3(256), 0,
                      stream>>>(x, xsum);
  }

  // 2) 5 layers, ping-pong activations; final layer writes d_out directly.
  const dim3 grid(NB / 16, 2);  // 128 batch tiles x 2 directions
  const dim3 block(32);

  lstm_layer_kernel<1><<<grid, block, 0, stream>>>(
      xsum, bufA, W_ih0, W_hh, b_ih, b_hh);

  const float* lin[4]  = {bufA, bufB, bufA, bufB};
  float*       lout[4] = {bufB, bufA, bufB, out};
  for (int l = 1; l < 5; ++l) {
    lstm_layer_kernel<22><<<grid, block, 0, stream>>>(
        lin[l - 1], lout[l - 1],
        W_ihr + (size_t)(l - 1) * 2 * GG * 22,
        W_hh  + (size_t)l * 2 * GG * HH,
        b_ih  + (size_t)l * 2 * GG,
        b_hh  + (size_t)l * 2 * GG);
  }
}